// LiftBlock_6390911336494
// MI455X (gfx1250) — compile-verified
//
#include <hip/hip_runtime.h>

typedef float v2f __attribute__((ext_vector_type(2)));
typedef float v8f __attribute__((ext_vector_type(8)));

#define N_NODES 100000
#define N_EDGES 1600000
#define R_DIM   6
#define CH_IN   8
#define CH_OUT  16
#define K_DIM   96          // R * C_IN * 2 (complex expanded along K)
#define OCOLS   32          // C_OUT * 2 (real | imag output columns)
#define N_TILES (N_NODES / 16)   // 6250, exact

// ---------------------------------------------------------------------------
// Phase 1: edge scatter.  thread = (edge, r).  16 hw global_atomic_add_f32
// into agg[dst*96 + r*16 + c*2 + {re,im}] -- 64 contiguous bytes per thread.
// agg (38.4 MB) is L2-resident (192 MB L2), so atomics resolve at L2 rate.
// Edge/stencil streams are read-once -> non-temporal loads so they don't
// compete with agg/x for L2 residency.
// ---------------------------------------------------------------------------
__global__ __launch_bounds__(256) void scatter_kernel(
    const float* __restrict__ x,
    const int*   __restrict__ edges,     // (E,2) int32
    const float* __restrict__ sten,      // (E,R,2) f32
    float*       __restrict__ agg)       // (N,96) f32
{
  int i = blockIdx.x * 256 + threadIdx.x;     // 0 .. E*R-1 (exact grid)
  int e = i / R_DIM;
  int r = i - e * R_DIM;
  if (e >= N_EDGES) return;

  // one NT b64 load per edge row (src,dst packed)
  long long ep = __builtin_nontemporal_load((const long long*)edges + e);
  int src = (int)(ep & 0xffffffffLL);
  int dst = (int)(ep >> 32);

  v2f st = __builtin_nontemporal_load((const v2f*)sten + (size_t)e * R_DIM + r);

  const float4* xp = (const float4*)(x + (size_t)src * CH_IN);   // L2-hot
  float4 x0 = xp[0];
  float4 x1 = xp[1];
  float xs[8] = {x0.x, x0.y, x0.z, x0.w, x1.x, x1.y, x1.z, x1.w};

  float* base = agg + (size_t)dst * K_DIM + r * (2 * CH_IN);
#pragma unroll
  for (int c = 0; c < CH_IN; ++c) {
    unsafeAtomicAdd(base + 2 * c,     st.x * xs[c]);   // real
    unsafeAtomicAdd(base + 2 * c + 1, st.y * xs[c]);   // imag
  }
}

// ---------------------------------------------------------------------------
// Phase 2: build real-expanded filter F, PAIR-PACKED for 8-byte LDS loads:
//   logical F[k][col], k = (r*8+c)*2 + p, col<16 real out / col>=16 imag out
//     F[(rc,0)][o]    =  w*cos   F[(rc,1)][o]    = -w*sin
//     F[(rc,0)][16+o] =  w*sin   F[(rc,1)][16+o] =  w*cos
//   stored as Fp[(k>>1)*64 + col*2 + (k&1)]  (pair of consecutive K per b64)
// ---------------------------------------------------------------------------
__global__ __launch_bounds__(256) void filter_kernel(
    const float* __restrict__ weight,    // (6,8,16)
    const float* __restrict__ offset,    // (8,16)
    float*       __restrict__ Fp)        // (48,32,2)
{
  int i = blockIdx.x * 256 + threadIdx.x;
  if (i >= K_DIM * OCOLS) return;
  int k   = i >> 5;
  int col = i & 31;
  int p   = k & 1;
  int rc  = k >> 1;
  int r   = rc >> 3;
  int c   = rc & 7;
  int o   = col & 15;

  float w  = weight[(r * CH_IN + c) * CH_OUT + o];
  float ph = offset[c * CH_OUT + o];
  float fre = w * cosf(ph);
  float fim = w * sinf(ph);

  float val;
  if (col < 16) val = (p == 0) ? fre : -fim;
  else          val = (p == 0) ? fim :  fre;

  Fp[(rc * OCOLS + col) * 2 + p] = val;
}

// ---------------------------------------------------------------------------
// Phase 3: 16-node tile GEMM via V_WMMA_F32_16X16X4_F32, K = 96.
// acc0 = real columns (o = 0..15), acc1 = imag columns.  Identical (m,n)->lane
// mapping in both accumulators => magnitude/ReLU epilogue is lane-local.
// Each B operand is a single contiguous ds_load_b64 (pair-packed layout).
// ---------------------------------------------------------------------------
__global__ __launch_bounds__(256) void gemm_kernel(
    const float* __restrict__ Fp,    // (48,32,2) pair-packed filter
    const float* __restrict__ agg,   // (N,96)
    const float* __restrict__ bias,  // (16,)
    float*       __restrict__ out)   // (N,16,2)
{
  __shared__ float Fl[K_DIM * OCOLS];           // 12 KB
  for (int i = threadIdx.x; i < K_DIM * OCOLS; i += 256) Fl[i] = Fp[i];
  __syncthreads();

  int wave = threadIdx.x >> 5;
  int lane = threadIdx.x & 31;
  int tile = blockIdx.x * 8 + wave;
  if (tile >= N_TILES) return;                  // wave-uniform

  int node_base = tile * 16;
  int mrow = lane & 15;                         // A row / B,C,D column
  int half = lane >> 4;                         // 0: K+0..1, 1: K+2..3
  const float* arow = agg + (size_t)(node_base + mrow) * K_DIM + half * 2;
  // LDS base for this lane: column slot + half-selected pair
  const float* fb0 = Fl + (size_t)(half * OCOLS + mrow) * 2;        // cols 0-15
  const float* fb1 = fb0 + 32;                                      // cols 16-31

  v8f acc0 = {0.f, 0.f, 0.f, 0.f, 0.f, 0.f, 0.f, 0.f};
  v8f acc1 = {0.f, 0.f, 0.f, 0.f, 0.f, 0.f, 0.f, 0.f};

#pragma unroll 4
  for (int k0 = 0; k0 < K_DIM; k0 += 4) {
    // A 16x4 f32: vgpr0 = K=k0 (lanes 0-15) / k0+2 (lanes 16-31), vgpr1 = +1
    v2f a  = *(const v2f*)(arow + k0);          // global b64, 8B aligned
    // B 4x16 f32, one ds_load_b64 each: {F[kk][col], F[kk+1][col]}
    v2f b0 = *(const v2f*)(fb0 + (k0 >> 1) * (OCOLS * 2));
    v2f b1 = *(const v2f*)(fb1 + (k0 >> 1) * (OCOLS * 2));
    acc0 = __builtin_amdgcn_wmma_f32_16x16x4_f32(
        false, a, false, b0, (short)0, acc0, false, false);
    acc1 = __builtin_amdgcn_wmma_f32_16x16x4_f32(
        false, a, false, b1, (short)0, acc1, false, false);
  }

  // Epilogue: mag/ReLU gate with single-instruction sqrt/rcp (TRANS ops),
  // coalesced non-temporal float2 {re,im} stores.
  float bn = bias[mrow];
  int   mhi = half * 8;
#pragma unroll
  for (int j = 0; j < 8; ++j) {
    int   m   = j + mhi;
    float re  = acc0[j];
    float im  = acc1[j];
    float mag = __builtin_amdgcn_sqrtf(re * re + im * im);   // v_sqrt_f32
    float sc  = fmaxf(mag + bn, 0.0f) *
                __builtin_amdgcn_rcpf(mag + 1e-8f);          // v_rcp_f32
    v2f v = { re * sc, im * sc };
    __builtin_nontemporal_store(
        v, (v2f*)(out + ((size_t)(node_base + m) * CH_OUT + mrow) * 2));
  }
}

// ---------------------------------------------------------------------------
extern "C" void kernel_launch(void* const* d_in, const int* in_sizes, int n_in,
                              void* d_out, int out_size, void* d_ws, size_t ws_size,
                              hipStream_t stream) {
  const float* x      = (const float*)d_in[0];
  const int*   edges  = (const int*)d_in[1];
  const float* sten   = (const float*)d_in[2];
  const float* weight = (const float*)d_in[3];
  const float* offset = (const float*)d_in[4];
  const float* bias   = (const float*)d_in[5];
  float*       out    = (float*)d_out;

  // workspace: agg (N*96 f32 = 38.4 MB, L2-resident) then pair-packed F
  float* agg = (float*)d_ws;
  float* Fp  = agg + (size_t)N_NODES * K_DIM;

  hipMemsetAsync(agg, 0, (size_t)N_NODES * K_DIM * sizeof(float), stream);

  filter_kernel<<<(K_DIM * OCOLS + 255) / 256, 256, 0, stream>>>(weight, offset, Fp);

  long long total = (long long)N_EDGES * R_DIM;            // 9.6M, /256 exact
  scatter_kernel<<<(int)((total + 255) / 256), 256, 0, stream>>>(x, edges, sten, agg);

  gemm_kernel<<<(N_TILES + 7) / 8, 256, 0, stream>>>(Fp, agg, bias, out);
}